// SelfAttention_38139309589081
// MI455X (gfx1250) — compile-verified
//
#include <hip/hip_runtime.h>
#include <hip/hip_bf16.h>

typedef __attribute__((ext_vector_type(2))) float v2f;
typedef __attribute__((ext_vector_type(8))) float v8f;

#define T_DIM 4096
#define IN_SZ 1024
#define OUT_SZ 1024
#define BLK 60
#define EPSF 1e-12f

// -------- block reduction helpers (256 threads) --------
__device__ __forceinline__ float block_reduce_sum(float v, float* sm) {
    const int tid = threadIdx.x;
    sm[tid] = v; __syncthreads();
    for (int s = 128; s > 0; s >>= 1) {
        if (tid < s) sm[tid] += sm[tid + s];
        __syncthreads();
    }
    float r = sm[0]; __syncthreads();
    return r;
}
__device__ __forceinline__ float block_reduce_max(float v, float* sm) {
    const int tid = threadIdx.x;
    sm[tid] = v; __syncthreads();
    for (int s = 128; s > 0; s >>= 1) {
        if (tid < s) sm[tid] = fmaxf(sm[tid], sm[tid + s]);
        __syncthreads();
    }
    float r = sm[0]; __syncthreads();
    return r;
}

// -------- generic f32 WMMA GEMM, register-blocked + double-buffered --------
// C[M,N] = A[M,K] * op(B);  op(B) = TB ? B[N,K]^T (NT) : B[K,N] (NN)
// Macro tile 128(M) x 64(N), KC = 32.  256 threads = 8 wave32 waves
// arranged 4(M) x 2(N); each wave computes a 2x2 grid of 16x16 WMMA
// accumulators (32x32 of C) -> each LDS fragment feeds two WMMAs.
// Both A and B tiles are stored [row][k] with 34-float row stride so every
// WMMA fragment is one aligned 8-byte LDS read (ds_load_b64, no fixup movs).
// LDS is double-buffered: chunk k+1 is staged while chunk k's WMMAs run;
// one barrier per chunk.  All M,N,K here are multiples of the tile sizes
// -> no bounds checks, EXEC stays all-ones around every v_wmma.
__global__ __launch_bounds__(256) void gemm_f32_wmma(
    const float* __restrict__ A, const float* __restrict__ B, float* __restrict__ C,
    int M, int N, int K, int lda, int ldb, int ldc, int TB)
{
    __shared__ float a_t[2][128][34];   // [buf][m][k]
    __shared__ float b_t[2][64][34];    // [buf][n][k]
    const int tid  = threadIdx.x;
    const int lane = tid & 31;
    const int wave = tid >> 5;
    const int mw = wave >> 1;            // 0..3  (M: 32-row strip)
    const int nw = wave & 1;             // 0..1  (N: 32-col strip)
    const int tileM = blockIdx.y * 128;
    const int tileN = blockIdx.x * 64;

    const int lm = lane & 15;            // M (A) / N (B) index in fragment
    const int kh = (lane >> 4) << 1;     // K half-select (ISA 16x16x4 layout)

    v8f acc00 = {}, acc01 = {}, acc10 = {}, acc11 = {};

    auto stage = [&](int bf, int k0) {
        // A tile 128x32, coalesced along k, contiguous LDS writes
#pragma unroll
        for (int t = 0; t < 16; ++t) {
            int li = tid + t * 256;
            int r = li >> 5, c = li & 31;
            a_t[bf][r][c] = A[(size_t)(tileM + r) * lda + (k0 + c)];
        }
        // B tile -> b_t[n][k]  (64 x 32)
        if (TB) {
#pragma unroll
            for (int t = 0; t < 8; ++t) {
                int li = tid + t * 256;
                int r = li >> 5, c = li & 31;       // r = n, c = k
                b_t[bf][r][c] = B[(size_t)(tileN + r) * ldb + (k0 + c)];
            }
        } else {
#pragma unroll
            for (int t = 0; t < 8; ++t) {
                int li = tid + t * 256;
                int r = li >> 6, c = li & 63;       // r = k, c = n
                b_t[bf][c][r] = B[(size_t)(k0 + r) * ldb + (tileN + c)];
            }
        }
    };

    stage(0, 0);
    __syncthreads();

    int buf = 0;
    for (int k0 = 0; k0 < K; k0 += 32) {
        if (k0 + 32 < K) stage(buf ^ 1, k0 + 32);          // overlap with WMMAs below
        if (k0 + 64 < K)                                   // cover chunk after next
            __builtin_prefetch(&A[(size_t)(tileM + (tid >> 1)) * lda + (k0 + 64) + ((tid & 1) << 4)], 0, 0);

        const int ar = mw * 32 + lm;
        const int bc = nw * 32 + lm;
#pragma unroll
        for (int kk = 0; kk < 32; kk += 4) {
            v2f a0 = *(const v2f*)&a_t[buf][ar][kk + kh];
            v2f a1 = *(const v2f*)&a_t[buf][ar + 16][kk + kh];
            v2f b0 = *(const v2f*)&b_t[buf][bc][kk + kh];
            v2f b1 = *(const v2f*)&b_t[buf][bc + 16][kk + kh];
            acc00 = __builtin_amdgcn_wmma_f32_16x16x4_f32(false, a0, false, b0, (short)0, acc00, false, false);
            acc01 = __builtin_amdgcn_wmma_f32_16x16x4_f32(false, a0, false, b1, (short)0, acc01, false, false);
            acc10 = __builtin_amdgcn_wmma_f32_16x16x4_f32(false, a1, false, b0, (short)0, acc10, false, false);
            acc11 = __builtin_amdgcn_wmma_f32_16x16x4_f32(false, a1, false, b1, (short)0, acc11, false, false);
        }
        __syncthreads();
        buf ^= 1;
    }
    // C/D layout: VGPR v, lanes 0-15 -> M=v, lanes 16-31 -> M=v+8; N = lane%16
    const int half = (lane >> 4) << 3;
    const int rbase = tileM + mw * 32 + half;
    const int cbase = tileN + nw * 32 + lm;
#pragma unroll
    for (int v = 0; v < 8; ++v) {
        C[(size_t)(rbase + v)      * ldc + cbase]      = acc00[v];
        C[(size_t)(rbase + v)      * ldc + cbase + 16] = acc01[v];
        C[(size_t)(rbase + 16 + v) * ldc + cbase]      = acc10[v];
        C[(size_t)(rbase + 16 + v) * ldc + cbase + 16] = acc11[v];
    }
}

// -------- x -> L2-normalized rows --------
__global__ __launch_bounds__(256) void rownorm(const float* __restrict__ x,
                                               float* __restrict__ xu) {
    __shared__ float sm[256];
    const int row = blockIdx.x, tid = threadIdx.x;
    const float* xr = x + (size_t)row * IN_SZ;
    float s = 0.f;
    for (int c = tid; c < IN_SZ; c += 256) { float v = xr[c]; s += v * v; }
    s = block_reduce_sum(s, sm);
    const float inv = 1.0f / fmaxf(sqrtf(s), EPSF);
    for (int c = tid; c < IN_SZ; c += 256)
        xu[(size_t)row * IN_SZ + c] = xr[c] * inv;
}

// -------- row softmax (in place) + normalized row entropy --------
// entropy_raw = (ln S - (sum z*e^z)/S) / ln(T),   z = e - max
__global__ __launch_bounds__(256) void softmax_ent(float* __restrict__ E,
                                                   float* __restrict__ ent_raw) {
    __shared__ float sm[256];
    const int row = blockIdx.x, tid = threadIdx.x;
    float* er = E + (size_t)row * T_DIM;
    float mx = -3.402823466e38f;
    for (int j = tid; j < T_DIM; j += 256) mx = fmaxf(mx, er[j]);
    mx = block_reduce_max(mx, sm);
    float s = 0.f, zs = 0.f;
    for (int j = tid; j < T_DIM; j += 256) {
        float z = er[j] - mx;
        float e = expf(z);
        er[j] = e;
        s += e; zs += z * e;
    }
    s  = block_reduce_sum(s, sm);
    zs = block_reduce_sum(zs, sm);
    const float inv = 1.0f / s;
    for (int j = tid; j < T_DIM; j += 256) er[j] *= inv;
    if (tid == 0)
        ent_raw[row] = (logf(s) - zs * inv) * (1.0f / logf((float)T_DIM));
}

// -------- L1 norm of entropy vector --------
__global__ __launch_bounds__(256) void l1sum(const float* __restrict__ ent,
                                             float* __restrict__ out) {
    __shared__ float sm[256];
    const int tid = threadIdx.x;
    float s = 0.f;
    for (int j = tid; j < T_DIM; j += 256) s += fabsf(ent[j]);
    s = block_reduce_sum(s, sm);
    if (tid == 0) out[0] = s;
}

// -------- s_all[c] = sum_rows xu[r][c] --------
__global__ __launch_bounds__(256) void colsum(const float* __restrict__ xu,
                                              float* __restrict__ sall) {
    const int c = blockIdx.x * 256 + threadIdx.x;
    if (c < IN_SZ) {
        float s = 0.f;
        for (int r = 0; r < T_DIM; ++r) s += xu[(size_t)r * IN_SZ + c];
        sall[c] = s;
    }
}

// -------- dep_factor without materializing the TxT Gram matrix --------
// num_i = (1 - sum_blk a) - (xu_i.P_i - sum_blk a*g)
// den_i = (T - nblk)      - (xu_i.s_all - sum_blk g)
__global__ __launch_bounds__(256) void dep_kernel(
    const float* __restrict__ Xu, const float* __restrict__ P,
    const float* __restrict__ sall, const float* __restrict__ A,
    float* __restrict__ dep)
{
    __shared__ float sm[256];
    const int i = blockIdx.x, tid = threadIdx.x;
    const float* xi = Xu + (size_t)i * IN_SZ;

    float dp = 0.f, ds = 0.f;
    for (int c = tid; c < IN_SZ; c += 256) {
        float xv = xi[c];
        dp += xv * P[(size_t)i * IN_SZ + c];
        ds += xv * sall[c];
    }
    dp = block_reduce_sum(dp, sm);
    ds = block_reduce_sum(ds, sm);

    const int b = i / BLK;
    const int j0 = b * BLK;
    const int j1 = (j0 + BLK < T_DIM) ? (j0 + BLK) : T_DIM;

    float sga = 0.f, sg = 0.f, sa = 0.f;   // meaningful in tid 0 only
    for (int j = j0; j < j1; ++j) {
        float gp = 0.f;
        const float* xj = Xu + (size_t)j * IN_SZ;
        for (int c = tid; c < IN_SZ; c += 256) gp += xi[c] * xj[c];
        gp = block_reduce_sum(gp, sm);
        if (tid == 0) {
            float a = A[(size_t)i * T_DIM + j];
            sga += gp * a; sg += gp; sa += a;
        }
    }
    if (tid == 0) {
        float num = (1.0f - sa) - (dp - sga);
        float den = (float)(T_DIM - (j1 - j0)) - (ds - sg);
        dep[i] = num / den;
    }
}

// -------- att_win = keep ? (att + dep[col]) : 0  -> d_out region 2 --------
__global__ __launch_bounds__(256) void build_attwin(const float* __restrict__ A,
                                                    const float* __restrict__ dep,
                                                    float* __restrict__ out2) {
    size_t idx = (size_t)blockIdx.x * 256 + threadIdx.x;
    int i = (int)(idx / T_DIM), j = (int)(idx % T_DIM);
    bool keep = (i / BLK) == (j / BLK);
    out2[idx] = keep ? (A[idx] + dep[j]) : 0.0f;
}

// -------- add entropy/dep columns of Wout: rank-2 update of y_out --------
__global__ __launch_bounds__(256) void epilogue(float* __restrict__ y_out,
                                                const float* __restrict__ ent_raw,
                                                const float* __restrict__ entsum,
                                                const float* __restrict__ dep,
                                                const float* __restrict__ Wout) {
    size_t idx = (size_t)blockIdx.x * 256 + threadIdx.x;
    int m = (int)(idx / IN_SZ), n = (int)(idx % IN_SZ);
    float es = fmaxf(entsum[0], EPSF);
    const size_t wrow = (size_t)n * (OUT_SZ + 2);
    y_out[idx] += (ent_raw[m] / es) * Wout[wrow + OUT_SZ]
                + dep[m] * Wout[wrow + OUT_SZ + 1];
}

extern "C" void kernel_launch(void* const* d_in, const int* in_sizes, int n_in,
                              void* d_out, int out_size, void* d_ws, size_t ws_size,
                              hipStream_t stream) {
    (void)in_sizes; (void)n_in; (void)out_size; (void)ws_size;
    const float* x    = (const float*)d_in[0];
    const float* Wk   = (const float*)d_in[1];
    const float* Wq   = (const float*)d_in[2];
    const float* Wv   = (const float*)d_in[3];
    const float* Wout = (const float*)d_in[4];

    float* out     = (float*)d_out;
    float* y_out   = out;                                  // [T, 1024]
    float* att_win = out + (size_t)T_DIM * IN_SZ;          // [T, T]

    // workspace layout (floats): ~134 MB total
    float* ws = (float*)d_ws;
    const size_t SZ = (size_t)T_DIM * IN_SZ;               // 4M floats
    float* Q   = ws;                                       // later reused as P
    float* Km  = ws + SZ;                                  // later reused as Y
    float* V   = ws + 2 * SZ;
    float* Xu  = ws + 3 * SZ;
    float* E   = ws + 4 * SZ;                              // [T, T] 16M floats
    float* ent_raw = E + (size_t)T_DIM * T_DIM;            // [T]
    float* dep     = ent_raw + T_DIM;                      // [T]
    float* sall    = dep + T_DIM;                          // [1024]
    float* entsum  = sall + IN_SZ;                         // [1]

    dim3 blk(256);
    dim3 g_kx(IN_SZ / 64, T_DIM / 128);                    // C = [4096 x 1024]
    dim3 g_tt(T_DIM / 64, T_DIM / 128);                    // C = [4096 x 4096]

    // 1. xu = normalize(x)
    rownorm<<<T_DIM, blk, 0, stream>>>(x, Xu);
    // 2-4. Q/K/V = x @ W^T   (NT)
    gemm_f32_wmma<<<g_kx, blk, 0, stream>>>(x, Wq, Q,  T_DIM, OUT_SZ, IN_SZ, IN_SZ, IN_SZ, OUT_SZ, 1);
    gemm_f32_wmma<<<g_kx, blk, 0, stream>>>(x, Wk, Km, T_DIM, OUT_SZ, IN_SZ, IN_SZ, IN_SZ, OUT_SZ, 1);
    gemm_f32_wmma<<<g_kx, blk, 0, stream>>>(x, Wv, V,  T_DIM, OUT_SZ, IN_SZ, IN_SZ, IN_SZ, OUT_SZ, 1);
    // 5. E = Q @ K^T   (NT)
    gemm_f32_wmma<<<g_tt, blk, 0, stream>>>(Q, Km, E, T_DIM, T_DIM, OUT_SZ, OUT_SZ, OUT_SZ, T_DIM, 1);
    // 6. A = softmax(E) in place, entropy_raw
    softmax_ent<<<T_DIM, blk, 0, stream>>>(E, ent_raw);
    // 7. entsum = sum |entropy_raw|
    l1sum<<<1, blk, 0, stream>>>(ent_raw, entsum);
    // 8. P = A @ Xu   (NN) -- reuse Q's space (Q dead after step 5)
    float* P = Q;
    gemm_f32_wmma<<<g_kx, blk, 0, stream>>>(E, Xu, P, T_DIM, IN_SZ, T_DIM, T_DIM, IN_SZ, IN_SZ, 0);
    // 9. s_all = column sums of Xu
    colsum<<<IN_SZ / 256, blk, 0, stream>>>(Xu, sall);
    // 10. dep_factor
    dep_kernel<<<T_DIM, blk, 0, stream>>>(Xu, P, sall, E, dep);
    // 11. att_win -> d_out region 2
    build_attwin<<<(unsigned)(((size_t)T_DIM * T_DIM) / 256), blk, 0, stream>>>(E, dep, att_win);
    // 12. Y = att_win @ V   (NN) -- reuse K's space
    float* Y = Km;
    gemm_f32_wmma<<<g_kx, blk, 0, stream>>>(att_win, V, Y, T_DIM, OUT_SZ, T_DIM, T_DIM, OUT_SZ, OUT_SZ, 0);
    // 13. y_out = Y @ Wout[:, :1024]^T   (NT, ldb = 1026)
    gemm_f32_wmma<<<g_kx, blk, 0, stream>>>(Y, Wout, y_out, T_DIM, IN_SZ, OUT_SZ, OUT_SZ, OUT_SZ + 2, IN_SZ, 1);
    // 14. += entropy/dep columns
    epilogue<<<(unsigned)(((size_t)T_DIM * IN_SZ) / 256), blk, 0, stream>>>(y_out, ent_raw, entsum, dep, Wout);
}